// PageRank_CPU_47519518163098
// MI455X (gfx1250) — compile-verified
//
#include <hip/hip_runtime.h>
#include <stdint.h>

#define DAMPING_F   0.85f
#define TOL_F       1e-7f
#define MAX_ITERS   20
#define BLOCK       256
#define TILE        (BLOCK * 4)   // 1024 edges per tile

// ---- CDNA5 async global->LDS path (ASYNCcnt-tracked) ----
#if defined(__has_builtin)
#  if __has_builtin(__builtin_amdgcn_global_load_async_to_lds_b128) && \
      __has_builtin(__builtin_amdgcn_s_wait_asynccnt)
#    define HAVE_ASYNC_LDS 1
#  endif
#endif
#ifndef HAVE_ASYNC_LDS
#  define HAVE_ASYNC_LDS 0
#endif

#if HAVE_ASYNC_LDS
// builtin signature (clang-verified): (v4i addrspace(1)*, v4i addrspace(3)*, imm off, imm cpol)
typedef int v4i_t __attribute__((vector_size(16)));
typedef __attribute__((address_space(1))) v4i_t* g_v4i_p;
typedef __attribute__((address_space(3))) v4i_t* l_v4i_p;
#endif

__device__ __forceinline__ float block_reduce_add(float v, float* sdata) {
    const int tid = threadIdx.x;
    sdata[tid] = v;
    __syncthreads();
    for (int s = BLOCK / 2; s > 0; s >>= 1) {
        if (tid < s) sdata[tid] += sdata[tid + s];
        __syncthreads();
    }
    return sdata[0];
}

// ---------------- setup kernels ----------------

__global__ void k_zero_deg(int* __restrict__ deg, int n) {
    int i = blockIdx.x * BLOCK + threadIdx.x;
    if (i < n) deg[i] = 0;
}

__global__ void k_zero_sc(float* __restrict__ sc) {
    int i = threadIdx.x;
    if (i < 128) sc[i] = 0.0f;
}

// degree histogram: 4 edges per thread, vector load + int atomics
__global__ void k_deg(const int* __restrict__ src, int* __restrict__ deg, int E) {
    int base = (blockIdx.x * BLOCK + threadIdx.x) * 4;
    if (base + 4 <= E) {
        int4 s = *(const int4*)(src + base);
        atomicAdd(&deg[s.x], 1);
        atomicAdd(&deg[s.y], 1);
        atomicAdd(&deg[s.z], 1);
        atomicAdd(&deg[s.w], 1);
    } else {
        for (int k = 0; k < 4; ++k) {
            int e = base + k;
            if (e < E) atomicAdd(&deg[src[e]], 1);
        }
    }
}

// convert deg (int) -> dsafe (float) in place; sign bit encodes "dangling"
// dsafe = -1.0 for deg==0 (divide by |.|==1, contribute to blind sum), else (float)deg
__global__ void k_degsafe(int* __restrict__ deg_in_dsafe_out, int n) {
    int i = blockIdx.x * BLOCK + threadIdx.x;
    if (i < n) {
        int d = deg_in_dsafe_out[i];
        float f = (d == 0) ? -1.0f : (float)d;
        ((float*)deg_in_dsafe_out)[i] = f;
    }
}

__global__ void k_suminit(const float* __restrict__ x, float* __restrict__ sum, int n) {
    __shared__ float sd[BLOCK];
    int i = blockIdx.x * BLOCK + threadIdx.x;
    float v = (i < n) ? x[i] : 0.0f;
    float s = block_reduce_add(v, sd);
    if (threadIdx.x == 0) atomicAdd(sum, s);
}

__global__ void k_initv(const float* __restrict__ x, const float* __restrict__ sum,
                        float* __restrict__ vcur, int n) {
    int i = blockIdx.x * BLOCK + threadIdx.x;
    if (i < n) vcur[i] = x[i] / sum[0];
}

// ---------------- per-iteration kernels ----------------

// temp = v/|dsafe|; v_new = 0; blind += sum(v where dsafe<0)
__global__ void k_prep(const float* __restrict__ vcur, const float* __restrict__ dsafe,
                       float* __restrict__ temp, float* __restrict__ vnew,
                       float* __restrict__ blind_t, int n) {
    __shared__ float sd[BLOCK];
    int i = blockIdx.x * BLOCK + threadIdx.x;
    float contrib = 0.0f;
    if (i < n) {
        float v  = vcur[i];
        float ds = dsafe[i];
        temp[i] = v / fabsf(ds);      // exact same division as reference
        vnew[i] = 0.0f;
        if (ds < 0.0f) contrib = v;   // dangling node
    }
    float s = block_reduce_add(contrib, sd);
    if (threadIdx.x == 0) atomicAdd(blind_t, s);
}

// hot kernel: edge scatter with double-buffered async global->LDS pipeline.
// Each wave owns its LDS slice -> only s_wait_asynccnt needed (no barrier);
// s_wait_asynccnt(2) waits only for the current tile (async loads complete
// in order), so the next tile streams while this one's gather/atomics run.
__global__ void k_scatter(const int* __restrict__ src, const int* __restrict__ tgt,
                          const float* __restrict__ temp, float* __restrict__ vnew,
                          int E, int ntiles) {
    const int t = threadIdx.x;
#if HAVE_ASYNC_LDS
    __shared__ int s_src[2][TILE];
    __shared__ int s_tgt[2][TILE];
    int tile = blockIdx.x;
    const int stride = gridDim.x;
    int buf = 0;
    if (tile < ntiles) {
        int base = tile * TILE + t * 4;
        __builtin_amdgcn_global_load_async_to_lds_b128(
            (g_v4i_p)(src + base), (l_v4i_p)(&s_src[0][t * 4]), 0, 0);
        __builtin_amdgcn_global_load_async_to_lds_b128(
            (g_v4i_p)(tgt + base), (l_v4i_p)(&s_tgt[0][t * 4]), 0, 0);
    }
    while (tile < ntiles) {
        int next = tile + stride;
        if (next < ntiles) {
            int nbase = next * TILE + t * 4;
            __builtin_amdgcn_global_load_async_to_lds_b128(
                (g_v4i_p)(src + nbase), (l_v4i_p)(&s_src[buf ^ 1][t * 4]), 0, 0);
            __builtin_amdgcn_global_load_async_to_lds_b128(
                (g_v4i_p)(tgt + nbase), (l_v4i_p)(&s_tgt[buf ^ 1][t * 4]), 0, 0);
            __builtin_amdgcn_s_wait_asynccnt(2);   // current tile landed
        } else {
            __builtin_amdgcn_s_wait_asynccnt(0);
        }
#pragma unroll
        for (int k = 0; k < 4; ++k) {
            int s = s_src[buf][t * 4 + k];
            int d = s_tgt[buf][t * 4 + k];
            atomicAdd(&vnew[d], temp[s]);          // L2-resident gather + atomic
        }
        buf ^= 1;
        tile = next;
    }
#else
    for (int tile = blockIdx.x; tile < ntiles; tile += gridDim.x) {
        int base = tile * TILE + t * 4;
        int4 s = *(const int4*)(src + base);
        int4 d = *(const int4*)(tgt + base);
        atomicAdd(&vnew[d.x], temp[s.x]);
        atomicAdd(&vnew[d.y], temp[s.y]);
        atomicAdd(&vnew[d.z], temp[s.z]);
        atomicAdd(&vnew[d.w], temp[s.w]);
    }
#endif
    // tail edges (E % TILE)
    for (int e = ntiles * TILE + blockIdx.x * BLOCK + t; e < E;
         e += gridDim.x * BLOCK)
        atomicAdd(&vnew[tgt[e]], temp[src[e]]);
}

// vn = v_new*0.85 + base + blind*scale ; diff += |vn - v_cur| ;
// commit folded in: v_cur = vn unless frozen or t >= iteration
__global__ void k_finalize(float* __restrict__ vcur, const float* __restrict__ vnew,
                           const float* __restrict__ blind_t, float* __restrict__ diff_t,
                           const int* __restrict__ done_t, const int* __restrict__ iter,
                           int t, float basev, float bscale, int n) {
    __shared__ float sd[BLOCK];
    int i = blockIdx.x * BLOCK + threadIdx.x;
    float ad = 0.0f;
    if (i < n) {
        float vn = vnew[i] * DAMPING_F + (basev + blind_t[0] * bscale);
        float vc = vcur[i];
        ad = fabsf(vn - vc);
        if (done_t[0] == 0 && t < iter[0]) vcur[i] = vn;
    }
    float s = block_reduce_add(ad, sd);
    if (threadIdx.x == 0) atomicAdd(diff_t, s);
}

// done[t+1] = done[t] | (diff[t] < TOL)
__global__ void k_done(const float* __restrict__ diff_t, const int* __restrict__ done_t,
                       int* __restrict__ done_t1) {
    if (threadIdx.x == 0 && blockIdx.x == 0)
        done_t1[0] = done_t[0] | ((diff_t[0] < TOL_F) ? 1 : 0);
}

__global__ void k_out(const float* __restrict__ vcur, float* __restrict__ out, int n) {
    int i = blockIdx.x * BLOCK + threadIdx.x;
    if (i < n) out[i] = vcur[i];
}

// ---------------- launch ----------------

extern "C" void kernel_launch(void* const* d_in, const int* in_sizes, int n_in,
                              void* d_out, int out_size, void* d_ws, size_t ws_size,
                              hipStream_t stream) {
    const int*   src  = (const int*)d_in[0];
    const int*   tgt  = (const int*)d_in[1];
    const float* init = (const float*)d_in[2];
    const int*   iter = (const int*)d_in[3];   // scalar on device
    // d_in[4] = vertex_num (== n per reference)

    const int E = in_sizes[0];
    const int n = in_sizes[2];

    char* ws = (char*)d_ws;
    int*   deg   = (int*)ws;                        // becomes float dsafe in place
    float* dsafe = (float*)ws;
    float* temp  = (float*)(ws + (size_t)n * 4);
    float* vcur  = (float*)(ws + (size_t)n * 8);
    float* vnew  = (float*)(ws + (size_t)n * 12);
    float* sc    = (float*)(ws + (size_t)n * 16);   // 128-float scalar block
    float* sum_init = sc;              // sc[0]
    float* blind    = sc + 8;          // 20 slots
    float* diffv    = sc + 40;         // 20 slots
    int*   done     = (int*)(sc + 72); // 21 slots; done[0]=0 after zeroing

    const int gN     = (n + BLOCK - 1) / BLOCK;
    const int gE4    = (E + BLOCK * 4 - 1) / (BLOCK * 4);
    const int ntiles = E / TILE;                     // full tiles for async path
    int gS = (ntiles + 1) / 2;                       // ~2 tiles per block (pipeline)
    if (gS < 1) gS = 1;
    const float basev  = 0.15f / (float)n;
    const float bscale = DAMPING_F / (float)n;

    k_zero_deg<<<gN, BLOCK, 0, stream>>>(deg, n);
    k_zero_sc<<<1, 128, 0, stream>>>(sc);
    k_deg<<<gE4, BLOCK, 0, stream>>>(src, deg, E);
    k_degsafe<<<gN, BLOCK, 0, stream>>>(deg, n);
    k_suminit<<<gN, BLOCK, 0, stream>>>(init, sum_init, n);
    k_initv<<<gN, BLOCK, 0, stream>>>(init, sum_init, vcur, n);

    for (int t = 0; t < MAX_ITERS; ++t) {
        k_prep<<<gN, BLOCK, 0, stream>>>(vcur, dsafe, temp, vnew, blind + t, n);
        k_scatter<<<gS, BLOCK, 0, stream>>>(src, tgt, temp, vnew, E, ntiles);
        k_finalize<<<gN, BLOCK, 0, stream>>>(vcur, vnew, blind + t, diffv + t,
                                             done + t, iter, t, basev, bscale, n);
        k_done<<<1, 1, 0, stream>>>(diffv + t, done + t, done + t + 1);
    }
    k_out<<<gN, BLOCK, 0, stream>>>(vcur, (float*)d_out, n);
}